// GE2ELoss_40956808134829
// MI455X (gfx1250) — compile-verified
//
#include <hip/hip_runtime.h>
#include <hip/hip_bf16.h>

typedef __attribute__((ext_vector_type(2))) float v2f;
typedef __attribute__((ext_vector_type(8))) float v8f;

#define DDIM 768
#define MUTT 40
#define NSPK 1024
#define ROWS 32            // rows per block (two 16-row WMMA tiles per wave)
#define LDA  770           // padded LDS stride: 768 % 64banks == 0 -> pad by 2 (keeps 8B align)

// ---------------- Kernel 1: per-row inverse L2 norms (1 wave per row) -------
__global__ __launch_bounds__(128)
void ge2e_row_norms(const float* __restrict__ x, float* __restrict__ rnorm) {
    const int row  = blockIdx.x * 4 + (threadIdx.x >> 5);
    const int lane = threadIdx.x & 31;
    const float4* p = (const float4*)(x + (size_t)row * DDIM);
    float ss = 0.0f;
#pragma unroll
    for (int i = 0; i < 6; ++i) {               // 6 * 32 lanes * float4 = 768
        float4 v = p[lane + 32 * i];
        ss = fmaf(v.x, v.x, fmaf(v.y, v.y, fmaf(v.z, v.z, fmaf(v.w, v.w, ss))));
    }
#pragma unroll
    for (int m = 16; m >= 1; m >>= 1) ss += __shfl_xor(ss, m, 32);
    if (lane == 0) rnorm[row] = 1.0f / fmaxf(sqrtf(ss), 1e-12f);
}

// ---------------- Kernel 2: inclusive centroids c_incl[n,d] -----------------
__global__ __launch_bounds__(256)
void ge2e_centroids(const float* __restrict__ x, const float* __restrict__ rnorm,
                    float* __restrict__ cincl) {
    __shared__ float rn[MUTT];
    const int n = blockIdx.x;
    if (threadIdx.x < MUTT) rn[threadIdx.x] = rnorm[n * MUTT + threadIdx.x];
    __syncthreads();
    for (int d = threadIdx.x; d < DDIM; d += 256) {
        const float* p = x + (size_t)n * MUTT * DDIM + d;
        float s = 0.0f;
#pragma unroll
        for (int m = 0; m < MUTT; ++m) s = fmaf(p[(size_t)m * DDIM], rn[m], s);
        cincl[(size_t)n * DDIM + d] = s * (1.0f / MUTT);
    }
}

// ------- Kernel 3: fused f32-WMMA GEMM + diagonal fix + logsumexp + mean ----
__global__ __launch_bounds__(256)
void ge2e_gemm_lse(const float* __restrict__ x, const float* __restrict__ rnorm,
                   const float* __restrict__ cincl, const float* __restrict__ wp,
                   const float* __restrict__ bp, float* __restrict__ out) {
    __shared__ float As[ROWS * LDA];     // normalized A stripe, 32 x 768 (padded)
    __shared__ float wsums[8][ROWS];     // per-wave partial sum-exp per row
    __shared__ float pickw[8];           // per-wave sum of picked logits

    const int tid   = threadIdx.x;
    const int wave  = tid >> 5;
    const int lane  = tid & 31;
    const int h     = lane >> 4;         // half-wave select
    const int l16   = lane & 15;
    const int rbase = blockIdx.x * ROWS;

    // Stage normalized A stripe into LDS (coalesced, normalize on the fly)
    for (int idx = tid; idx < ROWS * DDIM; idx += 256) {
        const int row = idx / DDIM;
        const int d   = idx - row * DDIM;
        As[row * LDA + d] = x[(size_t)(rbase + row) * DDIM + d] * rnorm[rbase + row];
    }
    __syncthreads();

    const float wscale = wp[0];
    const float bias   = bp[0];
    // All logits are w*sim+b with sim in [-1,1] (unit-vector dots), so this is
    // a guaranteed upper bound: exp(lg - bnd) can never overflow, and the
    // smallest term exp(-2|w|) stays far above f32 denormals -> no max tracking.
    const float bnd = fabsf(wscale) + bias;

    float ps[2][8];                      // per-lane partial sum-exp per row slot
#pragma unroll
    for (int s = 0; s < 2; ++s)
#pragma unroll
        for (int i = 0; i < 8; ++i) ps[s][i] = 0.0f;
    float pick = 0.0f;                   // per-lane sum of own-column logits

    const float* a0 = As + l16 * LDA;            // A row for sub-tile 0
    const float* a1 = As + (16 + l16) * LDA;     // A row for sub-tile 1

    // Each wave owns 128 columns: 8 col-tiles of 16, two 16-row tiles each.
    for (int t = 0; t < 8; ++t) {
        const int col = wave * 128 + t * 16 + l16;          // global column (speaker k)
        const float* bptr = cincl + (size_t)col * DDIM;     // B row (row-major c_incl)

        v8f acc[2] = {};
#pragma unroll 8
        for (int kb = 0; kb < DDIM; kb += 4) {
            const int d0 = kb + 2 * h;                       // f32 WMMA A/B lane layout
            v2f bv  = *(const v2f*)(bptr + d0);
            v2f av0 = *(const v2f*)(a0 + d0);
            v2f av1 = *(const v2f*)(a1 + d0);
            acc[0] = __builtin_amdgcn_wmma_f32_16x16x4_f32(false, av0, false, bv,
                                                           (short)0, acc[0], false, false);
            acc[1] = __builtin_amdgcn_wmma_f32_16x16x4_f32(false, av1, false, bv,
                                                           (short)0, acc[1], false, false);
        }

        // Epilogue: leave-one-out diagonal fix + bounded sum-exp (branchless)
#pragma unroll
        for (int s = 0; s < 2; ++s) {
#pragma unroll
            for (int i = 0; i < 8; ++i) {
                const int r = rbase + s * 16 + 8 * h + i;   // global (n,m) row
                float v = acc[s][i];
                const bool own = (col == r / MUTT);
                if (own) v = (MUTT * v - 1.0f) * (1.0f / (MUTT - 1));
                const float lg = fmaf(wscale, v, bias);
                pick += own ? lg : 0.0f;                    // cndmask, no exec churn
                ps[s][i] += __expf(lg - bnd);
            }
        }
    }

    // Cross-lane plain-sum butterflies (16-lane groups share a row per slot)
#pragma unroll
    for (int s = 0; s < 2; ++s)
#pragma unroll
        for (int i = 0; i < 8; ++i)
#pragma unroll
            for (int msk = 1; msk < 16; msk <<= 1)
                ps[s][i] += __shfl_xor(ps[s][i], msk, 32);
    // pick: full-wave sum
#pragma unroll
    for (int msk = 16; msk >= 1; msk >>= 1) pick += __shfl_xor(pick, msk, 32);

    if (l16 == 0) {
#pragma unroll
        for (int s = 0; s < 2; ++s)
#pragma unroll
            for (int i = 0; i < 8; ++i)
                wsums[wave][s * 16 + 8 * h + i] = ps[s][i];
    }
    if (lane == 0) pickw[wave] = pick;
    __syncthreads();

    // Wave 0: per-row total sum-exp -> lse; block-sum lse; subtract picked; 1 atomic
    if (wave == 0) {
        float ssum = 0.0f;
#pragma unroll
        for (int wv = 0; wv < 8; ++wv) ssum += wsums[wv][lane];   // 32 rows <-> 32 lanes
        float lse = bnd + __logf(ssum);
#pragma unroll
        for (int msk = 16; msk >= 1; msk >>= 1) lse += __shfl_xor(lse, msk, 32);
        if (lane == 0) {
            float psum = 0.0f;
#pragma unroll
            for (int wv = 0; wv < 8; ++wv) psum += pickw[wv];
            atomicAdd(out, (lse - psum) * (1.0f / (NSPK * MUTT)));
        }
    }
}

extern "C" void kernel_launch(void* const* d_in, const int* in_sizes, int n_in,
                              void* d_out, int out_size, void* d_ws, size_t ws_size,
                              hipStream_t stream) {
    const float* x = (const float*)d_in[0];
    const float* w = (const float*)d_in[1];
    const float* b = (const float*)d_in[2];
    float* out   = (float*)d_out;
    float* rnorm = (float*)d_ws;                 // 40960 floats
    float* cincl = rnorm + NSPK * MUTT;          // 1024*768 floats

    hipMemsetAsync(d_out, 0, sizeof(float), stream);
    ge2e_row_norms<<<(NSPK * MUTT) / 4, 128, 0, stream>>>(x, rnorm);
    ge2e_centroids<<<NSPK, 256, 0, stream>>>(x, rnorm, cincl);
    ge2e_gemm_lse<<<(NSPK * MUTT) / ROWS, 256, 0, stream>>>(x, rnorm, cincl, w, b, out);
}